// EMD_90855738179776
// MI455X (gfx1250) — compile-verified
//
#include <hip/hip_runtime.h>
#include <math.h>

typedef __attribute__((ext_vector_type(2))) float v2f;
typedef __attribute__((ext_vector_type(8))) float v8f;

#define BB 8
#define NN 2048
#define MM 2048
#define FEPS 1e-9f
#define LOG2E 1.4426950408889634f

// ---------------------------------------------------------------------------
// WMMA tile helper: D[M][N] = g*(-2*(p1[M] . p2[N]) + |p1[M]|^2)
// A rows packed g*(-2x,-2y,-2z,|p1|^2), B cols packed (x,y,z,1).
// Adding g*|p2[N]|^2 per lane afterwards gives g*d2 directly.
// ---------------------------------------------------------------------------
__device__ __forceinline__ v8f dist_tile(v2f A, v2f Bv) {
    v8f c = {};
    return __builtin_amdgcn_wmma_f32_16x16x4_f32(
        /*neg_a=*/false, A, /*neg_b=*/false, Bv,
        /*c_mod=*/(short)0, c, /*reuse_a=*/false, /*reuse_b=*/false);
}

// Build the per-lane A operand (pre-scaled by g) for the 16-row strip at
// (b, rowBase). Lane l and lane l+16 both reference row rowBase + (lane&15).
__device__ __forceinline__ v2f make_A(const float* xyz1, int b, int rowBase,
                                      int lane, float g) {
    int half = lane >> 4, l = lane & 15;
    const float* p = &xyz1[(size_t)(b * NN + rowBase + l) * 3];
    float x = p[0], y = p[1], z = p[2];
    float n1 = x * x + y * y + z * z;
    v2f A;
    A.x = g * (half ? (-2.f * z) : (-2.f * x));
    A.y = g * (half ? n1 : (-2.f * y));
    return A;
}

// ---------------------------------------------------------------------------
// init: rL = rR = 1 (n==m so multiL=multiR=1), colsum = 0, cost = 0
// ---------------------------------------------------------------------------
__global__ void __launch_bounds__(256) k_init(float* rL, float* rR,
                                              float* colsum, float* cost) {
    int i = blockIdx.x * 256 + threadIdx.x;
    if (i < BB * NN) { rL[i] = 1.f; rR[i] = 1.f; colsum[i] = 0.f; }
    if (i < BB) cost[i] = 0.f;
}

// ---------------------------------------------------------------------------
// Sweep 1: rowsum_n = sum_m exp(level*d2)*rR[m];  s[n] = rL[n]/(rowsum+eps)
// Constant c = level*log2e is folded into A and |p2|^2 (exp2(d + n2) form).
// ---------------------------------------------------------------------------
__global__ void __launch_bounds__(128) k_rowsum(const float* xyz1,
                                                const float* xyz2,
                                                const float* rL,
                                                const float* rR, float* s,
                                                float c) {
    __shared__ float sPts[MM * 4];
    __shared__ float sN2[MM];
    __shared__ float sRR[MM];
    int b = blockIdx.x >> 5, rb = blockIdx.x & 31, t = threadIdx.x;
    for (int i = t; i < MM; i += 128) {
        const float* p = &xyz2[(size_t)(b * MM + i) * 3];
        float x = p[0], y = p[1], z = p[2];
        sPts[i * 4 + 0] = x; sPts[i * 4 + 1] = y;
        sPts[i * 4 + 2] = z; sPts[i * 4 + 3] = 1.f;
        sN2[i] = c * (x * x + y * y + z * z);
        sRR[i] = rR[b * MM + i];
    }
    __syncthreads();
    int lane = t & 31, wave = t >> 5, half = lane >> 4, l = lane & 15;
    int rowBase = rb * 64 + wave * 16;
    v2f A = make_A(xyz1, b, rowBase, lane, c);
    float rowacc[8] = {0.f, 0.f, 0.f, 0.f, 0.f, 0.f, 0.f, 0.f};
    for (int ct = 0; ct < MM / 16; ++ct) {
        int col = ct * 16 + l;
        v2f Bv = *(const v2f*)&sPts[col * 4 + half * 2];
        float n2v = sN2[col];
        float rrv = sRR[col];
        v8f d = dist_tile(A, Bv);
#pragma unroll
        for (int r = 0; r < 8; ++r) {
            float e = __builtin_amdgcn_exp2f(d[r] + n2v);
            rowacc[r] += e * rrv;
        }
    }
#pragma unroll
    for (int r = 0; r < 8; ++r) {
        float v = rowacc[r];
        v += __shfl_xor(v, 1);
        v += __shfl_xor(v, 2);
        v += __shfl_xor(v, 4);
        v += __shfl_xor(v, 8);
        rowacc[r] = v;
    }
    if (l == 0) {
#pragma unroll
        for (int r = 0; r < 8; ++r) {
            int row = rowBase + r + half * 8;
            s[b * NN + row] = rL[b * NN + row] / (rowacc[r] + FEPS);
        }
    }
}

// ---------------------------------------------------------------------------
// Sweep 2: colacc[m] = sum_n s[n]*exp(level*d2)   (true ss = rR[m]*colacc[m])
// Same c-folding as sweep 1.
// ---------------------------------------------------------------------------
__global__ void __launch_bounds__(128) k_colsum(const float* xyz1,
                                                const float* xyz2,
                                                const float* s, float* colsum,
                                                float c) {
    __shared__ float sPts[MM * 4];
    __shared__ float sN2[MM];
    __shared__ float sCol[MM];
    int b = blockIdx.x >> 5, rb = blockIdx.x & 31, t = threadIdx.x;
    for (int i = t; i < MM; i += 128) {
        const float* p = &xyz2[(size_t)(b * MM + i) * 3];
        float x = p[0], y = p[1], z = p[2];
        sPts[i * 4 + 0] = x; sPts[i * 4 + 1] = y;
        sPts[i * 4 + 2] = z; sPts[i * 4 + 3] = 1.f;
        sN2[i] = c * (x * x + y * y + z * z);
        sCol[i] = 0.f;
    }
    __syncthreads();
    int lane = t & 31, wave = t >> 5, half = lane >> 4, l = lane & 15;
    int rowBase = rb * 64 + wave * 16;
    v2f A = make_A(xyz1, b, rowBase, lane, c);
    float sr[8];
#pragma unroll
    for (int r = 0; r < 8; ++r) sr[r] = s[b * NN + rowBase + r + half * 8];
    for (int ct = 0; ct < MM / 16; ++ct) {
        int col = ct * 16 + l;
        v2f Bv = *(const v2f*)&sPts[col * 4 + half * 2];
        float n2v = sN2[col];
        v8f d = dist_tile(A, Bv);
        float colp = 0.f;
#pragma unroll
        for (int r = 0; r < 8; ++r) {
            float e = __builtin_amdgcn_exp2f(d[r] + n2v);
            colp += sr[r] * e;
        }
        atomicAdd(&sCol[col], colp);
    }
    __syncthreads();
    for (int i = t; i < MM; i += 128) atomicAdd(&colsum[b * MM + i], sCol[i]);
}

// ---------------------------------------------------------------------------
// ratio: q[m] = rR*min(rR/(rR*colacc+eps),1); recycle colsum for sweep 3
// ---------------------------------------------------------------------------
__global__ void __launch_bounds__(256) k_ratio(const float* rR, float* colsum,
                                               float* q) {
    int i = blockIdx.x * 256 + threadIdx.x;
    if (i < BB * MM) {
        float rr = rR[i];
        float ss = rr * colsum[i];
        float ratio = fminf(rr / (ss + FEPS), 1.f);
        q[i] = rr * ratio;
        colsum[i] = 0.f;
    }
}

// ---------------------------------------------------------------------------
// Sweep 3: w = s[n]*exp(level*d2)*q[m]; cost += w*dist; rL -= rowsum(w);
//          colsum[m] = colsum_n(w) (for rR update)
// Needs raw d2 for dist=sqrt(d2), so A is unscaled here (g=1).
// ---------------------------------------------------------------------------
__global__ void __launch_bounds__(128) k_final(const float* xyz1,
                                               const float* xyz2,
                                               const float* s, const float* q,
                                               float* rL, float* colsum,
                                               float* cost, float c) {
    __shared__ float sPts[MM * 4];
    __shared__ float sN2[MM];
    __shared__ float sQ[MM];
    __shared__ float sCol[MM];
    int b = blockIdx.x >> 5, rb = blockIdx.x & 31, t = threadIdx.x;
    for (int i = t; i < MM; i += 128) {
        const float* p = &xyz2[(size_t)(b * MM + i) * 3];
        float x = p[0], y = p[1], z = p[2];
        sPts[i * 4 + 0] = x; sPts[i * 4 + 1] = y;
        sPts[i * 4 + 2] = z; sPts[i * 4 + 3] = 1.f;
        sN2[i] = x * x + y * y + z * z;
        sQ[i] = q[b * MM + i];
        sCol[i] = 0.f;
    }
    __syncthreads();
    int lane = t & 31, wave = t >> 5, half = lane >> 4, l = lane & 15;
    int rowBase = rb * 64 + wave * 16;
    v2f A = make_A(xyz1, b, rowBase, lane, 1.f);
    float sr[8];
#pragma unroll
    for (int r = 0; r < 8; ++r) sr[r] = s[b * NN + rowBase + r + half * 8];
    float rowW[8] = {0.f, 0.f, 0.f, 0.f, 0.f, 0.f, 0.f, 0.f};
    float costacc = 0.f;
    for (int ct = 0; ct < MM / 16; ++ct) {
        int col = ct * 16 + l;
        v2f Bv = *(const v2f*)&sPts[col * 4 + half * 2];
        float n2v = sN2[col];
        float qv = sQ[col];
        v8f d = dist_tile(A, Bv);
        float colp = 0.f;
#pragma unroll
        for (int r = 0; r < 8; ++r) {
            float d2 = d[r] + n2v;
            float e = __builtin_amdgcn_exp2f(c * d2);
            float w = sr[r] * e * qv;
            float dist = sqrtf(fmaxf(d2, 1e-20f));
            costacc += w * dist;
            rowW[r] += w;
            colp += w;
        }
        atomicAdd(&sCol[col], colp);
    }
#pragma unroll
    for (int r = 0; r < 8; ++r) {
        float v = rowW[r];
        v += __shfl_xor(v, 1);
        v += __shfl_xor(v, 2);
        v += __shfl_xor(v, 4);
        v += __shfl_xor(v, 8);
        rowW[r] = v;
    }
    if (l == 0) {
#pragma unroll
        for (int r = 0; r < 8; ++r) {
            int row = rowBase + r + half * 8;
            rL[b * NN + row] = fmaxf(rL[b * NN + row] - rowW[r], 0.f);
        }
    }
    costacc += __shfl_xor(costacc, 1);
    costacc += __shfl_xor(costacc, 2);
    costacc += __shfl_xor(costacc, 4);
    costacc += __shfl_xor(costacc, 8);
    costacc += __shfl_xor(costacc, 16);
    if (lane == 0) atomicAdd(&cost[b], costacc);
    __syncthreads();
    for (int i = t; i < MM; i += 128) atomicAdd(&colsum[b * MM + i], sCol[i]);
}

// ---------------------------------------------------------------------------
// update: rR = max(rR - colsum(w), 0); clear colsum for next iteration
// ---------------------------------------------------------------------------
__global__ void __launch_bounds__(256) k_update(float* rR, float* colsum) {
    int i = blockIdx.x * 256 + threadIdx.x;
    if (i < BB * MM) {
        rR[i] = fmaxf(rR[i] - colsum[i], 0.f);
        colsum[i] = 0.f;
    }
}

extern "C" void kernel_launch(void* const* d_in, const int* in_sizes, int n_in,
                              void* d_out, int out_size, void* d_ws,
                              size_t ws_size, hipStream_t stream) {
    const float* xyz1 = (const float*)d_in[0];  // [B,N,3]
    const float* xyz2 = (const float*)d_in[1];  // [B,M,3]
    float* cost = (float*)d_out;                // [B]
    float* ws = (float*)d_ws;                   // 5 * B*N floats = 320 KB
    float* rL = ws;
    float* rR = ws + BB * NN;
    float* s = ws + 2 * BB * NN;
    float* colsum = ws + 3 * BB * NN;
    float* q = ws + 4 * BB * NN;

    k_init<<<64, 256, 0, stream>>>(rL, rR, colsum, cost);

    // levels: -4^j for j = 7..-1, then 0
    const float levels[10] = {-16384.f, -4096.f, -1024.f, -256.f, -64.f,
                              -16.f,    -4.f,    -1.f,    -0.25f, 0.f};
    for (int it = 0; it < 10; ++it) {
        float c = levels[it] * LOG2E;  // exp(level*d2) = exp2(c*d2)
        k_rowsum<<<256, 128, 0, stream>>>(xyz1, xyz2, rL, rR, s, c);
        k_colsum<<<256, 128, 0, stream>>>(xyz1, xyz2, s, colsum, c);
        k_ratio<<<64, 256, 0, stream>>>(rR, colsum, q);
        k_final<<<256, 128, 0, stream>>>(xyz1, xyz2, s, q, rL, colsum, cost, c);
        k_update<<<64, 256, 0, stream>>>(rR, colsum);
    }
}